// SparseEmbedding_30279519437287
// MI455X (gfx1250) — compile-verified
//
#include <hip/hip_runtime.h>
#include <cstdint>

// ---------------------------------------------------------------------------
// Problem constants (from reference)
// ---------------------------------------------------------------------------
#define NROWS   204800          // 4096 * 50 gathered rows
#define DIM     64
#define NEMB    1000000
#define TABLE_F 64000000LL      // NEMB * DIM floats per state array
#define GSTEP   100

// d_out layout (flat float32, reference return order)
#define OFF_OUT 0LL
#define OFF_W   (204800LL * 64)          // 13,107,200
#define OFF_M   (OFF_W + TABLE_F)
#define OFF_V   (OFF_M + TABLE_F)
#define OFF_S   (OFF_V + TABLE_F)

// ---------------------------------------------------------------------------
// TDM availability probe (device pass only). ROCm 7.2 clang-22 exposes the
// 5-arg builtin; therock/clang-23 (ships the gfx1250 TDM header) uses 6 args.
// ---------------------------------------------------------------------------
#if defined(__HIP_DEVICE_COMPILE__) && defined(__gfx1250__) && \
    __has_builtin(__builtin_amdgcn_tensor_load_to_lds) &&       \
    __has_builtin(__builtin_amdgcn_tensor_store_from_lds) &&    \
    __has_builtin(__builtin_amdgcn_s_wait_tensorcnt)
#define USE_TDM 1
#if __has_include(<hip/amd_detail/amd_gfx1250_TDM.h>)
#define TDM_SIX_ARGS 1
#endif
#endif

typedef unsigned int u32x4 __attribute__((ext_vector_type(4)));
typedef int          i32x4 __attribute__((ext_vector_type(4)));
typedef int          i32x8 __attribute__((ext_vector_type(8)));
typedef float        f32x4 __attribute__((ext_vector_type(4)));

// 16,000 floats = 64,000 bytes per DMA tile; 64,000,000 / 16,000 = 4,000 tiles/array
#define CHUNK_F          16000
#define CHUNKS_PER_ARRAY 4000
#define COPY_BLOCKS      (3 * CHUNKS_PER_ARRAY)   // 12,000

#if defined(USE_TDM)
// D# group 0: [1:0]=count=1, [63:32]=lds_addr, [120:64]=global_addr, [127:126]=type(2)
__device__ __forceinline__ u32x4 tdm_group0(uint64_t gaddr, uint32_t lds_byte) {
  u32x4 g;
  g.x = 1u;                                             // count=1 (valid user D#)
  g.y = lds_byte;                                       // lds_addr (bytes)
  g.z = (uint32_t)gaddr;                                // global_addr[31:0]
  g.w = (uint32_t)((gaddr >> 32) & 0x1FFFFFFu) | (2u << 30); // addr[56:32] | type=2
  return g;
}

// D# group 1: data_size=4B, tensor = 1 row of CHUNK_F elems, tile = same row
__device__ __forceinline__ i32x8 tdm_group1() {
  i32x8 g;
  g[0] = 0x00020000;                    // workgroup_mask=0, data_size=2 (4 bytes)
  g[1] = (int)((unsigned)CHUNK_F << 16);// tensor_dim0[15:0] in [31:16]
  g[2] = (int)((((unsigned)CHUNK_F >> 16) & 0xFFFFu) | (1u << 16)); // dim0[31:16] | tensor_dim1=1
  g[3] = (int)((unsigned)CHUNK_F << 16);// tensor_dim1[31:16]=0 | tile_dim0=CHUNK_F
  g[4] = 1;                             // tile_dim1=1, tile_dim2=0
  g[5] = CHUNK_F;                       // tensor_dim0_stride[31:0]
  g[6] = 0;                             // stride0[47:32]=0 | tensor_dim1_stride[15:0]=0
  g[7] = 0;
  return g;
}
#endif

// ---------------------------------------------------------------------------
// Bulk copy: weight/exp_avgs/exp_avg_sqs -> d_out stale copies.
// TDM path: two DMA descriptors per 64KB tile staged through LDS.
// Fallback: nontemporal 128-bit vector streaming copy.
// ---------------------------------------------------------------------------
__global__ __launch_bounds__(32) void bulk_copy_kernel(
    const float* __restrict__ w, const float* __restrict__ m,
    const float* __restrict__ v, float* __restrict__ outbase) {
#if defined(USE_TDM)
  __shared__ float lbuf[CHUNK_F];
  const unsigned b     = blockIdx.x;
  const unsigned arr   = b / CHUNKS_PER_ARRAY;
  const unsigned chunk = b % CHUNKS_PER_ARRAY;
  const float* src = (arr == 0) ? w : (arr == 1) ? m : v;
  float*       dst = outbase + ((arr == 0) ? OFF_W : (arr == 1) ? OFF_M : OFF_V);
  const uint64_t ga_src = (uint64_t)(uintptr_t)(src + (uint64_t)chunk * CHUNK_F);
  const uint64_t ga_dst = (uint64_t)(uintptr_t)(dst + (uint64_t)chunk * CHUNK_F);
  const uint32_t lds_off = (uint32_t)(uintptr_t)lbuf;   // flat LDS addr truncates to LDS offset

  const i32x8 g1 = tdm_group1();
  const i32x4 z4 = {0, 0, 0, 0};
  const u32x4 g0l = tdm_group0(ga_src, lds_off);
  const u32x4 g0s = tdm_group0(ga_dst, lds_off);
#if defined(TDM_SIX_ARGS)
  const i32x8 z8 = {0, 0, 0, 0, 0, 0, 0, 0};
  __builtin_amdgcn_tensor_load_to_lds(g0l, g1, z4, z4, z8, 0);
  __builtin_amdgcn_s_wait_tensorcnt(0);
  __builtin_amdgcn_tensor_store_from_lds(g0s, g1, z4, z4, z8, 0);
  __builtin_amdgcn_s_wait_tensorcnt(0);
#else
  __builtin_amdgcn_tensor_load_to_lds(g0l, g1, z4, z4, 0);
  __builtin_amdgcn_s_wait_tensorcnt(0);
  __builtin_amdgcn_tensor_store_from_lds(g0s, g1, z4, z4, 0);
  __builtin_amdgcn_s_wait_tensorcnt(0);
#endif
#else
  // Fallback: grid-stride nontemporal b128 streaming copy of all 3 arrays.
  const long long q_per_arr = TABLE_F / 4;        // float4 per array
  const long long total4    = 3 * q_per_arr;      // 48M float4
  const long long stride    = (long long)gridDim.x * blockDim.x;
  for (long long i = (long long)blockIdx.x * blockDim.x + threadIdx.x;
       i < total4; i += stride) {
    const long long arr = i / q_per_arr;
    const long long j   = i - arr * q_per_arr;
    const f32x4* src = (arr == 0) ? (const f32x4*)w
                     : (arr == 1) ? (const f32x4*)m : (const f32x4*)v;
    f32x4* dst = (f32x4*)(outbase + ((arr == 0) ? OFF_W : (arr == 1) ? OFF_M : OFF_V));
    f32x4 val = __builtin_nontemporal_load(src + j);
    __builtin_nontemporal_store(val, dst + j);
  }
#endif
}

// ---------------------------------------------------------------------------
// step (int32) -> d_out float copy (d_out is homogeneous fp32)
// ---------------------------------------------------------------------------
__global__ __launch_bounds__(256) void step_out_kernel(
    const int* __restrict__ step, float* __restrict__ dst, long long n) {
  const long long stride = (long long)gridDim.x * blockDim.x;
  for (long long i = (long long)blockIdx.x * blockDim.x + threadIdx.x;
       i < n; i += stride) {
    dst[i] = (float)step[i];
  }
}

// ---------------------------------------------------------------------------
// One wave32 per gathered row: gather w/m/v (float2 per lane = coalesced
// 256B row), apply lazy-Adam decay, scatter to out + updated state copies.
// Duplicate indices write identical values (sp comes from the unmodified
// input step array), matching the reference's .set semantics.
// ---------------------------------------------------------------------------
__global__ __launch_bounds__(256) void adam_update_kernel(
    const int* __restrict__ indices, const float* __restrict__ W,
    const float* __restrict__ M, const float* __restrict__ V,
    const int* __restrict__ step, float* __restrict__ out) {
  const int gid  = blockIdx.x * blockDim.x + threadIdx.x;
  const int row  = gid >> 5;
  const int lane = gid & 31;
  if (row >= NROWS) return;

  const int id = indices[row];                  // uniform within the wave
  const long long base = (long long)id * DIM;
  const float2* wp = (const float2*)(W + base);
  const float2* mp = (const float2*)(M + base);
  const float2* vp = (const float2*)(V + base);
  __builtin_prefetch(wp + lane, 0, 0);          // global_prefetch_b8
  __builtin_prefetch(mp + lane, 0, 0);
  __builtin_prefetch(vp + lane, 0, 0);

  const float sp = (float)(GSTEP - step[id]);   // steps skipped, 1..100
  // t = beta1/sqrt(beta2); precomputed logs (constant-folded)
  const float LN_T  = -0.104860267f;            // ln(0.9) - 0.5*ln(0.999)
  const float T_M1  = -0.099549668f;            // t - 1
  const float LN_B1 = -0.105360516f;            // ln(0.9)
  const float LN_B2 = -0.001000500f;            // ln(0.999)
  const float decay = (__expf(sp * LN_T) - 1.0f) / T_M1;
  const float c     = 1.0e-3f * decay;          // LR * decay
  const float b1p   = __expf(sp * LN_B1);
  const float b2p   = __expf(sp * LN_B2);

  const float2 wv = wp[lane];
  const float2 mv = mp[lane];
  const float2 vv = vp[lane];
  float2 wn, mn, vn;
  wn.x = wv.x - c * (mv.x / (sqrtf(vv.x) + 1e-8f));
  wn.y = wv.y - c * (mv.y / (sqrtf(vv.y) + 1e-8f));
  mn.x = mv.x * b1p; mn.y = mv.y * b1p;
  vn.x = vv.x * b2p; vn.y = vv.y * b2p;

  ((float2*)(out + OFF_OUT + (long long)row * DIM))[lane] = wn;  // gathered output
  ((float2*)(out + OFF_W + base))[lane] = wn;                    // new_weight
  ((float2*)(out + OFF_M + base))[lane] = mn;                    // new_exp_avgs
  ((float2*)(out + OFF_V + base))[lane] = vn;                    // new_exp_avg_sqs
  if (lane == 0) out[OFF_S + id] = (float)GSTEP;                 // new_step
}

// ---------------------------------------------------------------------------
// Launch: copy (TDM DMA) -> step copy -> scatter-update. Same stream, so the
// scatter is ordered after the stale-copy pass.
// ---------------------------------------------------------------------------
extern "C" void kernel_launch(void* const* d_in, const int* in_sizes, int n_in,
                              void* d_out, int out_size, void* d_ws, size_t ws_size,
                              hipStream_t stream) {
  (void)in_sizes; (void)n_in; (void)out_size; (void)d_ws; (void)ws_size;
  const int*   indices = (const int*)d_in[0];
  const float* weight  = (const float*)d_in[1];
  const float* m       = (const float*)d_in[2];
  const float* v       = (const float*)d_in[3];
  const int*   step    = (const int*)d_in[4];
  float* out = (float*)d_out;

  bulk_copy_kernel<<<COPY_BLOCKS, 32, 0, stream>>>(weight, m, v, out);
  step_out_kernel<<<512, 256, 0, stream>>>(step, out + OFF_S, (long long)NEMB);
  adam_update_kernel<<<(NROWS * 32 + 255) / 256, 256, 0, stream>>>(
      indices, weight, m, v, step, out);
}